// PRT_89266600280419
// MI455X (gfx1250) — compile-verified
//
#include <hip/hip_runtime.h>
#include <hip/hip_bf16.h>

typedef float v2f __attribute__((ext_vector_type(2)));
typedef float v8f __attribute__((ext_vector_type(8)));

#define PRT_S 256
#define PRT_K 16
#define PRT_W 0.04908738521234052f   // 4*pi / 256
#define WPB   8                      // waves (rays) per block
#define ROWST 17                     // LDS row stride in floats (bank-conflict free)
#define WLDS  (32 * ROWST)           // floats of LDS per wave

__global__ __launch_bounds__(256)
void prt_sh_kernel(const float* __restrict__ vis,
                   const float* __restrict__ phi,
                   const float* __restrict__ theta,
                   float* __restrict__ out, int R)
{
  __shared__ float lds[WPB * WLDS];
  const int lane = threadIdx.x & 31;
  const int wid  = threadIdx.x >> 5;
  const int ray0 = blockIdx.x * WPB + wid;
  const int ray  = (ray0 < R) ? ray0 : (R - 1);   // wave-uniform clamp (no divergence)

  const size_t base = (size_t)ray * PRT_S;
  const float4* th4 = reinterpret_cast<const float4*>(theta + base) + lane * 2;
  const float4* ph4 = reinterpret_cast<const float4*>(phi   + base) + lane * 2;
  const float4* pv4 = reinterpret_cast<const float4*>(vis   + base) + lane * 2;

  float acc[16];
  #pragma unroll
  for (int k = 0; k < 16; ++k) acc[k] = 0.0f;

  #pragma unroll
  for (int q = 0; q < 2; ++q) {
    const float4 T = th4[q];
    const float4 P = ph4[q];
    const float4 V = pv4[q];
    const float ta[4] = {T.x, T.y, T.z, T.w};
    const float pa[4] = {P.x, P.y, P.z, P.w};
    const float va[4] = {V.x, V.y, V.z, V.w};
    #pragma unroll
    for (int j = 0; j < 4; ++j) {
      const float v   = va[j];
      const float x   = __cosf(ta[j]);                    // cos(theta)
      const float s   = sqrtf((1.0f + x) * (1.0f - x));   // sin(theta) >= 0 (matches ref)
      const float sp  = __sinf(pa[j]);
      const float cp  = __cosf(pa[j]);
      const float s2p = 2.0f * sp * cp;                   // sin(2phi)
      const float c2p = fmaf(-2.0f * sp, sp, 1.0f);       // cos(2phi)
      const float s3p = fmaf(s2p, cp,  c2p * sp);         // sin(3phi)
      const float c3p = fmaf(c2p, cp, -s2p * sp);         // cos(3phi)
      const float x2  = x * x;
      const float ss  = s * s;
      const float s3  = ss * s;
      const float xs  = x * s;
      const float xs2 = x * ss;
      const float f5  = fmaf(5.0f, x2, -1.0f);            // 5x^2 - 1

      acc[0]  = fmaf( 0.28209479f,                          v, acc[0]);
      acc[1]  = fmaf(-0.48860252f * (s * sp),               v, acc[1]);
      acc[2]  = fmaf( 0.48860252f * x,                      v, acc[2]);
      acc[3]  = fmaf(-0.48860252f * (s * cp),               v, acc[3]);
      acc[4]  = fmaf( 0.54627422f * (ss * s2p),             v, acc[4]);
      acc[5]  = fmaf(-1.09254843f * (xs * sp),              v, acc[5]);
      acc[6]  = fmaf( 0.31539157f * fmaf(3.0f, x2, -1.0f),  v, acc[6]);
      acc[7]  = fmaf(-1.09254843f * (xs * cp),              v, acc[7]);
      acc[8]  = fmaf( 0.54627422f * (ss * c2p),             v, acc[8]);
      acc[9]  = fmaf(-0.59004359f * (s3 * s3p),             v, acc[9]);
      acc[10] = fmaf( 2.89061144f * (xs2 * s2p),            v, acc[10]);
      acc[11] = fmaf(-0.45704579f * (f5 * (s * sp)),        v, acc[11]);
      acc[12] = fmaf( 0.37317633f * (x * fmaf(5.0f, x2, -3.0f)), v, acc[12]);
      acc[13] = fmaf(-0.45704579f * (f5 * (s * cp)),        v, acc[13]);
      acc[14] = fmaf( 2.89061144f * (xs2 * c2p),            v, acc[14]);
      acc[15] = fmaf(-0.59004359f * (s3 * c3p),             v, acc[15]);
    }
  }

  // Transpose per-lane accumulators (harmonic k lives in register space) into
  // lane space through LDS, then finish the 32-lane reduction with one
  // f32-exact WMMA: D = ones(16x4) x B + 0 => every row of D = sum_K B[K][n].
  float* my = &lds[wid * WLDS];
  #pragma unroll
  for (int k = 0; k < 16; ++k) my[lane * ROWST + k] = acc[k];
  __syncthreads();

  const int half = lane >> 4;   // which half-wave
  const int col  = lane & 15;   // harmonic column this lane reduces
  float b0 = 0.0f, b1 = 0.0f;
  #pragma unroll
  for (int i = 0; i < 8; ++i) {
    b0 += my[(half * 8 + i) * ROWST + col];        // rows  0..15 -> K slots of VGPR0
    b1 += my[((2 + half) * 8 + i) * ROWST + col];  // rows 16..31 -> K slots of VGPR1
  }

  v2f A; A[0] = 1.0f; A[1] = 1.0f;
  v2f B; B[0] = b0;   B[1] = b1;
  v8f C = {0.f, 0.f, 0.f, 0.f, 0.f, 0.f, 0.f, 0.f};
  v8f D = __builtin_amdgcn_wmma_f32_16x16x4_f32(false, A, false, B,
                                                (short)0, C, false, false);

  float r = D[0] * PRT_W;                 // row M=0, col n = lane (lanes 0-15)
  r = fminf(fmaxf(r, -2.0f), 2.0f);
  if (lane < 16 && ray0 < R)
    out[(size_t)ray0 * PRT_K + lane] = r;
}

extern "C" void kernel_launch(void* const* d_in, const int* in_sizes, int n_in,
                              void* d_out, int out_size, void* d_ws, size_t ws_size,
                              hipStream_t stream) {
  const float* vis   = (const float*)d_in[0];   // p_vis  [R, S, 1]
  const float* phi   = (const float*)d_in[1];   // phi    [R, S]
  const float* theta = (const float*)d_in[2];   // theta  [R, S]
  float* out = (float*)d_out;                   // [R, 16]
  const int R = in_sizes[1] / PRT_S;            // 32768
  const int blocks = (R + WPB - 1) / WPB;
  hipLaunchKernelGGL(prt_sh_kernel, dim3(blocks), dim3(256), 0, stream,
                     vis, phi, theta, out, R);
}